// SinkhornLoss_19275813224505
// MI455X (gfx1250) — compile-verified
//
#include <hip/hip_runtime.h>

typedef __attribute__((ext_vector_type(2))) float v2f;
typedef __attribute__((ext_vector_type(8))) float v8f;

#define NPTS 8192
#define KDIM 32
#define LOG_N 9.010913347279288f  // log(8192)
#define NITER 10

// ---------------------------------------------------------------------------
// ws layout (floats): [0,8192) xx  [8192,16384) yy  [16384,24576) f  [24576,32768) g
// ---------------------------------------------------------------------------

__global__ __launch_bounds__(256) void sq_norm_kernel(const float* __restrict__ X,
                                                      float* __restrict__ out) {
  int i = blockIdx.x * blockDim.x + threadIdx.x;
  if (i >= NPTS) return;
  const float4* p = (const float4*)(X + (size_t)i * KDIM);
  float s = 0.f;
#pragma unroll
  for (int k = 0; k < KDIM / 4; ++k) {
    float4 v = p[k];
    s += v.x * v.x + v.y * v.y + v.z * v.z + v.w * v.w;
  }
  out[i] = s;
}

__global__ __launch_bounds__(256) void init_kernel(float* __restrict__ f,
                                                   float* __restrict__ g,
                                                   float* __restrict__ out) {
  int i = blockIdx.x * blockDim.x + threadIdx.x;
  if (i < NPTS) {
    f[i] = 0.f;
    g[i] = 0.f;
  }
  if (i == 0) out[0] = 0.f;
}

// One Sinkhorn half-update:
//   outdual[i] = -log(N) + a_sq[i] - log( sum_j exp((dual[j]-b_sq[j]) + 2*dot(A_i,B_j)) )
// Block = 32-row strip of A (two 16-row WMMA strips); 8 waves partition the
// 512 column tiles of B; two independent WMMA accumulation chains per tile.
__global__ __launch_bounds__(256) void sinkhorn_pass_kernel(
    const float* __restrict__ A, const float* __restrict__ B,
    const float* __restrict__ a_sq, const float* __restrict__ b_sq,
    const float* __restrict__ dual, float* __restrict__ outdual) {
  __shared__ float s_db[NPTS];      // dual[j] - b_sq[j]
  __shared__ float s_part[8 * 32];  // per-wave partial row sums (32 rows)

  for (int i = threadIdx.x; i < NPTS; i += 256) s_db[i] = dual[i] - b_sq[i];
  __syncthreads();

  const int wave = threadIdx.x >> 5;
  const int lane = threadIdx.x & 31;
  const int l16 = lane & 15;
  const int hi = lane >> 4;  // which 16-lane half
  const int row_base = blockIdx.x * 32;

  // Preload A fragments for both 16-row strips: K=32 -> 8 slabs of K=4.
  // f32 16x4 A layout: lane l holds row l%16, K = 2*(l/16) + {0,1} within slab.
  v2f a0[8], a1[8];
  const float* arow0 = A + (size_t)(row_base + l16) * KDIM;
  const float* arow1 = arow0 + 16 * KDIM;
#pragma unroll
  for (int k = 0; k < 8; ++k) {
    a0[k] = *(const v2f*)(arow0 + k * 4 + 2 * hi);
    a1[k] = *(const v2f*)(arow1 + k * 4 + 2 * hi);
  }

  float acc0[8], acc1[8];
#pragma unroll
  for (int v = 0; v < 8; ++v) {
    acc0[v] = 0.f;
    acc1[v] = 0.f;
  }

  for (int jt = wave; jt < NPTS / 16; jt += 8) {
    const int col_base = jt * 16;
    const float* brow = B + (size_t)(col_base + l16) * KDIM;
    v2f b[8];
#pragma unroll
    for (int k = 0; k < 8; ++k)
      b[k] = *(const v2f*)(brow + k * 4 + 2 * hi);

    v8f c0 = {}, c1 = {};  // dots for strip 0 / strip 1 (independent chains)
#pragma unroll
    for (int k = 0; k < 8; ++k) {
      c0 = __builtin_amdgcn_wmma_f32_16x16x4_f32(false, a0[k], false, b[k],
                                                 (short)0, c0, false, false);
      c1 = __builtin_amdgcn_wmma_f32_16x16x4_f32(false, a1[k], false, b[k],
                                                 (short)0, c1, false, false);
    }

    const float db = s_db[col_base + l16];
#pragma unroll
    for (int v = 0; v < 8; ++v) {
      acc0[v] += __expf(fmaf(2.f, c0[v], db));
      acc1[v] += __expf(fmaf(2.f, c1[v], db));
    }
  }

  // Reduce over the 16 lanes of each half-wave (same row, different columns).
#pragma unroll
  for (int v = 0; v < 8; ++v) {
    float s0 = acc0[v], s1 = acc1[v];
#pragma unroll
    for (int off = 1; off < 16; off <<= 1) {
      s0 += __shfl_xor(s0, off, 32);
      s1 += __shfl_xor(s1, off, 32);
    }
    if (l16 == 0) {
      s_part[wave * 32 + v + 8 * hi] = s0;       // strip 0: rows base+0..15
      s_part[wave * 32 + 16 + v + 8 * hi] = s1;  // strip 1: rows base+16..31
    }
  }
  __syncthreads();

  if (threadIdx.x < 32) {
    float s = 0.f;
#pragma unroll
    for (int w = 0; w < 8; ++w) s += s_part[w * 32 + threadIdx.x];
    const int r = row_base + threadIdx.x;
    outdual[r] = -LOG_N + a_sq[r] - __logf(s);
  }
}

// loss += sum_ij C_ij * exp(f_i + g_j - C_ij), C_ij = a_sq_i + b_sq_j - 2*dot
__global__ __launch_bounds__(256) void sinkhorn_loss_kernel(
    const float* __restrict__ A, const float* __restrict__ B,
    const float* __restrict__ a_sq, const float* __restrict__ b_sq,
    const float* __restrict__ f, const float* __restrict__ g,
    float* __restrict__ out) {
  __shared__ float s_gb[NPTS];  // g[j] - b_sq[j]
  __shared__ float s_part[8];

  for (int i = threadIdx.x; i < NPTS; i += 256) s_gb[i] = g[i] - b_sq[i];
  __syncthreads();

  const int wave = threadIdx.x >> 5;
  const int lane = threadIdx.x & 31;
  const int l16 = lane & 15;
  const int hi = lane >> 4;
  const int row_base = blockIdx.x * 32;

  v2f a0[8], a1[8];
  const float* arow0 = A + (size_t)(row_base + l16) * KDIM;
  const float* arow1 = arow0 + 16 * KDIM;
#pragma unroll
  for (int k = 0; k < 8; ++k) {
    a0[k] = *(const v2f*)(arow0 + k * 4 + 2 * hi);
    a1[k] = *(const v2f*)(arow1 + k * 4 + 2 * hi);
  }

  float aa0[8], fa0[8], aa1[8], fa1[8];
#pragma unroll
  for (int v = 0; v < 8; ++v) {
    const int r0 = row_base + v + 8 * hi;  // row held by this lane in slot v
    const int r1 = r0 + 16;
    aa0[v] = a_sq[r0];
    fa0[v] = f[r0] - aa0[v];
    aa1[v] = a_sq[r1];
    fa1[v] = f[r1] - aa1[v];
  }

  float acc = 0.f;
  for (int jt = wave; jt < NPTS / 16; jt += 8) {
    const int col_base = jt * 16;
    const float* brow = B + (size_t)(col_base + l16) * KDIM;
    v2f b[8];
#pragma unroll
    for (int k = 0; k < 8; ++k)
      b[k] = *(const v2f*)(brow + k * 4 + 2 * hi);

    v8f c0 = {}, c1 = {};
#pragma unroll
    for (int k = 0; k < 8; ++k) {
      c0 = __builtin_amdgcn_wmma_f32_16x16x4_f32(false, a0[k], false, b[k],
                                                 (short)0, c0, false, false);
      c1 = __builtin_amdgcn_wmma_f32_16x16x4_f32(false, a1[k], false, b[k],
                                                 (short)0, c1, false, false);
    }

    const float gb = s_gb[col_base + l16];
    const float bq = b_sq[col_base + l16];
#pragma unroll
    for (int v = 0; v < 8; ++v) {
      const float td0 = 2.f * c0[v];
      const float td1 = 2.f * c1[v];
      acc = fmaf(aa0[v] + bq - td0, __expf(fa0[v] + gb + td0), acc);
      acc = fmaf(aa1[v] + bq - td1, __expf(fa1[v] + gb + td1), acc);
    }
  }

  // Full-wave reduce, then cross-wave, then one atomic per block.
#pragma unroll
  for (int off = 1; off < 32; off <<= 1) acc += __shfl_xor(acc, off, 32);
  if (lane == 0) s_part[wave] = acc;
  __syncthreads();
  if (threadIdx.x == 0) {
    float s = 0.f;
#pragma unroll
    for (int w = 0; w < 8; ++w) s += s_part[w];
    atomicAdd(out, s);
  }
}

extern "C" void kernel_launch(void* const* d_in, const int* in_sizes, int n_in,
                              void* d_out, int out_size, void* d_ws, size_t ws_size,
                              hipStream_t stream) {
  const float* x = (const float*)d_in[0];
  const float* y = (const float*)d_in[1];
  float* out = (float*)d_out;
  float* ws = (float*)d_ws;
  float* xx = ws;
  float* yy = ws + NPTS;
  float* f = ws + 2 * NPTS;
  float* g = ws + 3 * NPTS;

  init_kernel<<<NPTS / 256, 256, 0, stream>>>(f, g, out);
  sq_norm_kernel<<<NPTS / 256, 256, 0, stream>>>(x, xx);
  sq_norm_kernel<<<NPTS / 256, 256, 0, stream>>>(y, yy);

  for (int it = 0; it < NITER; ++it) {
    // f-update uses g; g-update uses the fresh f (matches reference ordering)
    sinkhorn_pass_kernel<<<NPTS / 32, 256, 0, stream>>>(x, y, xx, yy, g, f);
    sinkhorn_pass_kernel<<<NPTS / 32, 256, 0, stream>>>(y, x, yy, xx, f, g);
  }

  sinkhorn_loss_kernel<<<NPTS / 32, 256, 0, stream>>>(x, y, xx, yy, f, g, out);
}